// _BaseTransducer_72713796321439
// MI455X (gfx1250) — compile-verified
//
#include <hip/hip_runtime.h>

typedef __attribute__((ext_vector_type(2))) float v2f;
typedef __attribute__((ext_vector_type(4))) float v4f;
typedef __attribute__((ext_vector_type(8))) float v8f;

#define B_   8
#define M_   256
#define N_   64
#define D_   512
#define C_   1024
#define SROWS_ (B_*M_)          // 2048 speech rows
#define ROWS_  (B_*M_ + B_*N_)  // 2560 total rows

// ---------------------------------------------------------------------------
// Kernel 1: Y[r,c] = sum_d X[r,d] * W[c,d]   (X = [speech_flat ; text_flat])
// One wave32 computes a 32x64 output tile with V_WMMA_F32_16X16X4_F32.
//
// fp32 WMMA fragment layout (ISA 7.12.2):
//   A (16x4,MxK): lane l -> row m = l&15, VGPR0/1 = K = 2*(l>>4) + {0,1}
//   B (4x16,KxN): mirrored -> lane l -> col n = l&15, K = 2*(l>>4) + {0,1}
// Both fragments are contiguous float2 loads from row-major X and row-major W
// (B = W^T, so consecutive K in a column of B = consecutive elements of a
// row of W). C/D (16x16): VGPR v, lanes0-15 -> M=v, lanes16-31 -> M=v+8.
// ---------------------------------------------------------------------------
__global__ __launch_bounds__(256) void proj_wmma_f32(
    const float* __restrict__ speech, const float* __restrict__ text,
    const float* __restrict__ W, float* __restrict__ Y)
{
    const int wave = threadIdx.x >> 5;
    const int lane = threadIdx.x & 31;
    const int tile = blockIdx.x * 8 + wave;     // 0..1279
    const int rt   = tile >> 4;                 // 80 row-tiles of 32 rows
    const int ct   = tile & 15;                 // 16 col-tiles of 64 cols
    const int r0   = rt * 32;
    const int half = lane >> 4;                 // 0 or 1
    const int lid  = lane & 15;
    const int ksel = half * 2;                  // K sub-offset within frag

    const float* src;
    int srow;
    if (r0 < SROWS_) { src = speech; srow = r0; }
    else             { src = text;   srow = r0 - SROWS_; }

    const float* a0p = src + (size_t)(srow +      lid) * D_ + ksel;
    const float* a1p = src + (size_t)(srow + 16 + lid) * D_ + ksel;
    const int c0 = ct * 64;
    const float* b0p = W + (size_t)(c0 +  0 + lid) * D_ + ksel;
    const float* b1p = W + (size_t)(c0 + 16 + lid) * D_ + ksel;
    const float* b2p = W + (size_t)(c0 + 32 + lid) * D_ + ksel;
    const float* b3p = W + (size_t)(c0 + 48 + lid) * D_ + ksel;

    v8f acc00 = {}, acc01 = {}, acc02 = {}, acc03 = {};
    v8f acc10 = {}, acc11 = {}, acc12 = {}, acc13 = {};

#pragma unroll 4
    for (int k = 0; k < D_; k += 4) {
        v2f a0 = *(const v2f*)(a0p + k);
        v2f a1 = *(const v2f*)(a1p + k);
        v2f b0 = *(const v2f*)(b0p + k);
        v2f b1 = *(const v2f*)(b1p + k);
        v2f b2 = *(const v2f*)(b2p + k);
        v2f b3 = *(const v2f*)(b3p + k);
        acc00 = __builtin_amdgcn_wmma_f32_16x16x4_f32(false, a0, false, b0, (short)0, acc00, false, false);
        acc01 = __builtin_amdgcn_wmma_f32_16x16x4_f32(false, a0, false, b1, (short)0, acc01, false, false);
        acc02 = __builtin_amdgcn_wmma_f32_16x16x4_f32(false, a0, false, b2, (short)0, acc02, false, false);
        acc03 = __builtin_amdgcn_wmma_f32_16x16x4_f32(false, a0, false, b3, (short)0, acc03, false, false);
        acc10 = __builtin_amdgcn_wmma_f32_16x16x4_f32(false, a1, false, b0, (short)0, acc10, false, false);
        acc11 = __builtin_amdgcn_wmma_f32_16x16x4_f32(false, a1, false, b1, (short)0, acc11, false, false);
        acc12 = __builtin_amdgcn_wmma_f32_16x16x4_f32(false, a1, false, b2, (short)0, acc12, false, false);
        acc13 = __builtin_amdgcn_wmma_f32_16x16x4_f32(false, a1, false, b3, (short)0, acc13, false, false);
    }

    const int col = c0 + lid;
    const int row = r0 + half * 8;
#pragma unroll
    for (int v = 0; v < 8; ++v) {
        float* y0 = Y + (size_t)(row + v) * C_ + col;
        y0[ 0] = acc00[v];  y0[16] = acc01[v];  y0[32] = acc02[v];  y0[48] = acc03[v];
        float* y1 = Y + (size_t)(row + 16 + v) * C_ + col;
        y1[ 0] = acc10[v];  y1[16] = acc11[v];  y1[32] = acc12[v];  y1[48] = acc13[v];
    }
}

// ---------------------------------------------------------------------------
// Kernel 2: out[b,m,n,c] = S[b,m,c] + T[b,n,c] + bias[c]
// Block = 256 threads covering all 1024 c's as float4; 8x8 (m,n) tile per
// block so each S/T row load is reused 8x from registers. Output is written
// once and never re-read (537 MB >> 192 MB L2) -> non-temporal b128 stores.
// ---------------------------------------------------------------------------
__global__ __launch_bounds__(256) void joint_bcast_add(
    const float* __restrict__ Y, const float* __restrict__ bias,
    float* __restrict__ out)
{
    const float* S = Y;                           // [B*M, C]
    const float* T = Y + (size_t)SROWS_ * C_;     // [B*N, C]
    const int b  = blockIdx.y;
    const int m0 = (blockIdx.x >> 3) * 8;
    const int n0 = (blockIdx.x &  7) * 8;
    const int c  = threadIdx.x * 4;

    const v4f bb = *(const v4f*)(bias + c);

    v4f sv[8], tv[8];
#pragma unroll
    for (int i = 0; i < 8; ++i)
        sv[i] = *(const v4f*)(S + (size_t)(b * M_ + m0 + i) * C_ + c) + bb;
#pragma unroll
    for (int j = 0; j < 8; ++j)
        tv[j] = *(const v4f*)(T + (size_t)(b * N_ + n0 + j) * C_ + c);

#pragma unroll
    for (int i = 0; i < 8; ++i) {
        float* orow = out + ((size_t)(b * M_ + m0 + i) * N_ + n0) * C_ + c;
#pragma unroll
        for (int j = 0; j < 8; ++j) {
            v4f r = sv[i] + tv[j];
            __builtin_nontemporal_store(r, (v4f*)(orow + (size_t)j * C_));
        }
    }
}

// ---------------------------------------------------------------------------
// Kernel 3: speech_len[b] = sum(speech_mask[b,:]), text_len[b] = sum(text_mask)
// Tiny (16 ints of output) -> one wave, serial per-row sums.
// ---------------------------------------------------------------------------
__global__ void lengths_k(const unsigned char* __restrict__ sm,
                          const unsigned char* __restrict__ tm,
                          int* __restrict__ lens)
{
    const int t = threadIdx.x;
    if (t < B_) {
        int s = 0;
        for (int i = 0; i < M_; ++i) s += sm[t * M_ + i] ? 1 : 0;
        lens[t] = s;
    } else if (t < 2 * B_) {
        const int bb = t - B_;
        int s = 0;
        for (int i = 0; i < N_; ++i) s += tm[bb * N_ + i] ? 1 : 0;
        lens[B_ + bb] = s;
    }
}

extern "C" void kernel_launch(void* const* d_in, const int* in_sizes, int n_in,
                              void* d_out, int out_size, void* d_ws, size_t ws_size,
                              hipStream_t stream)
{
    const float*         speech = (const float*)d_in[0];
    const unsigned char* smask  = (const unsigned char*)d_in[1];
    const float*         text   = (const float*)d_in[2];
    const unsigned char* tmask  = (const unsigned char*)d_in[3];
    const float*         W      = (const float*)d_in[4];
    const float*         bias   = (const float*)d_in[5];

    float* out = (float*)d_out;
    float* Y   = (float*)d_ws;   // 2560*1024 f32 = 10 MB scratch (S then T)

    // 1) S/T projection: 1280 wave-tiles, 8 waves/block -> 160 blocks.
    proj_wmma_f32<<<dim3(160), dim3(256), 0, stream>>>(speech, text, W, Y);

    // 2) Broadcast add + bias, streaming NT stores of the 537 MB result.
    joint_bcast_add<<<dim3((M_ / 8) * (N_ / 8), B_), dim3(256), 0, stream>>>(Y, bias, out);

    // 3) Lengths appended after the result tensor (int32 bit patterns).
    int* lens = (int*)d_out + (size_t)B_ * M_ * N_ * C_;
    lengths_k<<<dim3(1), dim3(32), 0, stream>>>(smask, tmask, lens);
}